// SetOfOrthonormalTransforms_82806969467548
// MI455X (gfx1250) — compile-verified
//
#include <hip/hip_runtime.h>
#include <hip/hip_bf16.h>

typedef float v2f __attribute__((ext_vector_type(2)));
typedef float v8f __attribute__((ext_vector_type(8)));

#define M_BATCH 1024
#define NDIMS   32
#define NANG    496
#define NSAMP   2048

// ---------------------------------------------------------------------------
// Kernel 1: build R[m] = diag(mus[m]) * prod(Givens rotations(angles[m]))
// One wave (32 lanes) per batch. Lane = column index j. v[i] = mat[i][j].
// Rotation (t,b): vt' = c*vt - s*vb ; vb' = s*vt + c*vb  (algebraic form of ref)
// Loop fully unrolled so row indices are compile-time constants -> registers.
// ---------------------------------------------------------------------------
__global__ void __launch_bounds__(256)
gen_R_kernel(const float* __restrict__ angles, const float* __restrict__ mus,
             float* __restrict__ R) {
    const int wave = blockIdx.x * 8 + (threadIdx.x >> 5);
    const int lane = threadIdx.x & 31;

    const float* a = angles + wave * NANG;

    float v[NDIMS];
#pragma unroll
    for (int i = 0; i < NDIMS; ++i) v[i] = (i == lane) ? 1.0f : 0.0f;

    int k = 0;
#pragma unroll
    for (int t = 0; t < NDIMS - 1; ++t) {
#pragma unroll
        for (int b = t + 1; b < NDIMS; ++b) {
            const float ang = a[k++];
            const float s = __sinf(ang);
            const float c = __cosf(ang);
            const float vt = v[t];
            const float vb = v[b];
            v[t] = fmaf(c, vt, -s * vb);
            v[b] = fmaf(s, vt,  c * vb);
        }
    }

    const float* mu = mus + wave * NDIMS;
    float* r = R + wave * NDIMS * NDIMS;
#pragma unroll
    for (int i = 0; i < NDIMS; ++i) {
        r[i * NDIMS + lane] = mu[i] * v[i];
    }
}

// ---------------------------------------------------------------------------
// Kernel 2: out[m] = R[m] (32x32) @ x[m] (32x2048) via V_WMMA_F32_16X16X4_F32.
// One wave computes a 32x64 output tile: 2 row-halves x 4 col-tiles of 16x16.
// K=32 consumed as 8 chained x4-K WMMA steps.
//
// Fragment layouts (CDNA5 ISA):
//   A 16x4:  lanes 0-15 = M, VGPR0:K=0|K=2 (lane halves), VGPR1:K=1|K=3
//   B 4x16:  VGPR0: row K=0 (lanes 0-15) | K=2 (lanes 16-31); VGPR1: K=1 | K=3
//   C 16x16: VGPR r: M=r (lanes 0-15) | M=r+8 (lanes 16-31), N=lane&15
// ---------------------------------------------------------------------------
__global__ void __launch_bounds__(256)
gemm_R_x_kernel(const float* __restrict__ R, const float* __restrict__ X,
                float* __restrict__ Y) {
    const int wave = blockIdx.x * 8 + (threadIdx.x >> 5);
    const int lane = threadIdx.x & 31;

    const int m  = wave >> 5;          // batch index
    const int n0 = (wave & 31) << 6;   // 64-column tile origin

    const float* r = R + m * NDIMS * NDIMS;
    const float* x = X + (size_t)m * NDIMS * NSAMP;
    float*       y = Y + (size_t)m * NDIMS * NSAMP;

    const int half = lane >> 4;   // 0: K even pair, 1: K odd pair (A/B); M vs M+8 (C)
    const int l    = lane & 15;

    v8f acc[2][4];
#pragma unroll
    for (int h = 0; h < 2; ++h)
#pragma unroll
        for (int nt = 0; nt < 4; ++nt)
#pragma unroll
            for (int e = 0; e < 8; ++e) acc[h][nt][e] = 0.0f;

#pragma unroll
    for (int kc = 0; kc < 8; ++kc) {
        const int k0 = kc * 4 + 2 * half;   // this lane's first K element

        // A fragments: rows 0-15 (h=0) and rows 16-31 (h=1) of R
        v2f afrag[2];
        afrag[0].x = r[l * NDIMS + k0];
        afrag[0].y = r[l * NDIMS + k0 + 1];
        afrag[1].x = r[(16 + l) * NDIMS + k0];
        afrag[1].y = r[(16 + l) * NDIMS + k0 + 1];

#pragma unroll
        for (int nt = 0; nt < 4; ++nt) {
            const int col = n0 + nt * 16 + l;
            v2f bfrag;
            bfrag.x = x[(size_t)k0 * NSAMP + col];
            bfrag.y = x[(size_t)(k0 + 1) * NSAMP + col];

            acc[0][nt] = __builtin_amdgcn_wmma_f32_16x16x4_f32(
                false, afrag[0], false, bfrag, (short)0, acc[0][nt], false, false);
            acc[1][nt] = __builtin_amdgcn_wmma_f32_16x16x4_f32(
                false, afrag[1], false, bfrag, (short)0, acc[1][nt], false, false);
        }
    }

    // Store: acc[h][nt] element e -> row 16*h + e + 8*half, col n0 + nt*16 + l
#pragma unroll
    for (int h = 0; h < 2; ++h) {
#pragma unroll
        for (int nt = 0; nt < 4; ++nt) {
            const int col = n0 + nt * 16 + l;
#pragma unroll
            for (int e = 0; e < 8; ++e) {
                const int row = 16 * h + e + 8 * half;
                y[(size_t)row * NSAMP + col] = acc[h][nt][e];
            }
        }
    }
}

extern "C" void kernel_launch(void* const* d_in, const int* in_sizes, int n_in,
                              void* d_out, int out_size, void* d_ws, size_t ws_size,
                              hipStream_t stream) {
    const float* x      = (const float*)d_in[0];  // (1024, 32, 2048)
    const float* angles = (const float*)d_in[1];  // (1024, 496)
    const float* mus    = (const float*)d_in[2];  // (1024, 32)
    float* out = (float*)d_out;                   // (1024, 32, 2048)
    float* R   = (float*)d_ws;                    // 1024 * 32 * 32 floats = 4 MB

    // Kernel 1: 1024 waves, 8 waves/block -> 128 blocks
    gen_R_kernel<<<M_BATCH / 8, 256, 0, stream>>>(angles, mus, R);

    // Kernel 2: 1024 batches * 32 waves/batch = 32768 waves -> 4096 blocks
    gemm_R_x_kernel<<<(M_BATCH * 32) / 8, 256, 0, stream>>>(R, x, out);
}